// Attention2D_30399778521616
// MI455X (gfx1250) — compile-verified
//
#include <hip/hip_runtime.h>

typedef __attribute__((ext_vector_type(2))) float v2f;
typedef __attribute__((ext_vector_type(8))) float v8f;

#define T_LEN 1024
#define D_DIM 128
#define NROWS (2 * T_LEN)          // B*T = 2048
#define QSCALE 0.8838834764831844f // 1 / (sqrt(2) * 0.8)

__device__ __forceinline__ v8f wmma_f32_16x16x4(v2f a, v2f b, v8f c) {
    // D = A(16x4,f32) * B(4x16,f32) + C(16x16,f32)
    return __builtin_amdgcn_wmma_f32_16x16x4_f32(
        /*neg_a=*/false, a, /*neg_b=*/false, b,
        /*c_mod=*/(short)0, c, /*reuse_a=*/false, /*reuse_b=*/false);
}

// ---------------------------------------------------------------------------
// Kernel 1: fused QKV projection.  Y = x @ W^T for W in {Wq, Wk, Wv}.
// One wave computes one 16x16 tile of all three outputs (A fragment reused).
// grid: 128 blocks * 8 waves = 1024 waves = 128 Mtiles * 8 Ntiles.
// ---------------------------------------------------------------------------
__global__ __launch_bounds__(256)
void qkv_kernel(const float* __restrict__ x,
                const float* __restrict__ Wq,
                const float* __restrict__ Wk,
                const float* __restrict__ Wv,
                float* __restrict__ Q,
                float* __restrict__ K,
                float* __restrict__ V) {
    const int lane  = threadIdx.x & 31;
    const int wid   = (blockIdx.x * blockDim.x + threadIdx.x) >> 5;
    const int mtile = wid >> 3;   // 0..127
    const int ntile = wid & 7;    // 0..7
    const int lrow  = lane & 15;
    const int lhalf = lane >> 4;  // 0 or 1

    const float* xp  = x  + (size_t)(mtile * 16 + lrow) * D_DIM + 2 * lhalf;
    const float* wqp = Wq + (size_t)(ntile * 16 + lrow) * D_DIM + 2 * lhalf;
    const float* wkp = Wk + (size_t)(ntile * 16 + lrow) * D_DIM + 2 * lhalf;
    const float* wvp = Wv + (size_t)(ntile * 16 + lrow) * D_DIM + 2 * lhalf;

    v8f cq = {}, ck = {}, cv = {};
#pragma unroll 4
    for (int k = 0; k < D_DIM; k += 4) {
        v2f a  = *(const v2f*)(xp + k);
        v2f bq = *(const v2f*)(wqp + k);
        v2f bk = *(const v2f*)(wkp + k);
        v2f bv = *(const v2f*)(wvp + k);
        cq = wmma_f32_16x16x4(a, bq, cq);
        ck = wmma_f32_16x16x4(a, bk, ck);
        cv = wmma_f32_16x16x4(a, bv, cv);
    }

    const int row0 = mtile * 16 + 8 * lhalf;   // C/D layout: M = r + 8*(lane>=16)
    const int col  = ntile * 16 + lrow;        // N = lane % 16
#pragma unroll
    for (int r = 0; r < 8; ++r) {
        const size_t o = (size_t)(row0 + r) * D_DIM + col;
        Q[o] = cq[r];
        K[o] = ck[r];
        V[o] = cv[r];
    }
}

// ---------------------------------------------------------------------------
// Kernel 2: streaming (flash-style) causal attention, dh = 2.
// One block per (b,h); 8 waves; each wave owns q-tiles qt = wave, wave+8, ...
// Score tile 16x16 = ONE v_wmma_f32_16x16x4_f32 (dh padded 2 -> 4 with zeros).
// Online softmax with intra-16-lane shfl_xor reductions (wave32 halves match
// the WMMA C-matrix row split M = r + 8*(lane/16)).
// ---------------------------------------------------------------------------
__global__ __launch_bounds__(256)
void attn_kernel(const float* __restrict__ Q,
                 const float* __restrict__ Km,
                 const float* __restrict__ Vm,
                 float* __restrict__ O) {
    const int lane  = threadIdx.x & 31;
    const int wave  = threadIdx.x >> 5;  // 0..7
    const int bh    = blockIdx.x;        // 0..127
    const int b     = bh >> 6;
    const int h     = bh & 63;
    const int lrow  = lane & 15;
    const int lhalf = lane >> 4;

    const size_t base = (size_t)b * T_LEN * D_DIM + 2 * h;
    const float* Qp = Q  + base;
    const float* Kp = Km + base;
    const float* Vp = Vm + base;
    float*       Op = O  + base;

    for (int qt = wave; qt < T_LEN / 16; qt += 8) {
        // A fragment: row M = lane%16, K = 2*lhalf + {0,1}; dh=2 -> upper half zero.
        v2f afrag;
        {
            const float* qr = Qp + (size_t)(qt * 16 + lrow) * D_DIM;
            const float q0 = qr[0] * QSCALE;
            const float q1 = qr[1] * QSCALE;
            afrag.x = (lhalf == 0) ? q0 : 0.0f;
            afrag.y = (lhalf == 0) ? q1 : 0.0f;
        }

        float mrow[8], lsum[8], ox[8], oy[8];
#pragma unroll
        for (int r = 0; r < 8; ++r) {
            mrow[r] = -3.0e38f; lsum[r] = 0.0f; ox[r] = 0.0f; oy[r] = 0.0f;
        }

        for (int kt = 0; kt <= qt; ++kt) {
            // B fragment: col N = lane%16 is K-row (kt*16+N); K-dim 0,1 in lower half.
            v2f bfrag;
            {
                const float* kr = Kp + (size_t)(kt * 16 + lrow) * D_DIM;
                const float k0 = kr[0];
                const float k1 = kr[1];
                bfrag.x = (lhalf == 0) ? k0 : 0.0f;
                bfrag.y = (lhalf == 0) ? k1 : 0.0f;
            }
            v8f cz = {};
            v8f s  = wmma_f32_16x16x4(afrag, bfrag, cz);

            // V row for this lane's key column (same for both halves).
            const float* vr = Vp + (size_t)(kt * 16 + lrow) * D_DIM;
            const float v0 = vr[0];
            const float v1 = vr[1];
            const bool diag = (kt == qt);

#pragma unroll
            for (int r = 0; r < 8; ++r) {
                const int qloc = r + 8 * lhalf;        // local query row 0..15
                float sv = s[r];
                sv = (diag && (lrow > qloc)) ? -3.0e38f : sv;  // strict causal mask
                // row max over the 16 key lanes of this half
                float mx = sv;
                mx = fmaxf(mx, __shfl_xor(mx, 1, 32));
                mx = fmaxf(mx, __shfl_xor(mx, 2, 32));
                mx = fmaxf(mx, __shfl_xor(mx, 4, 32));
                mx = fmaxf(mx, __shfl_xor(mx, 8, 32));
                const float mnew  = fmaxf(mrow[r], mx);
                const float alpha = __expf(mrow[r] - mnew);
                const float p     = __expf(sv - mnew);
                lsum[r] = lsum[r] * alpha + p;      // per-lane partial sum
                ox[r]   = ox[r]   * alpha + p * v0; // per-lane partial PV
                oy[r]   = oy[r]   * alpha + p * v1;
                mrow[r] = mnew;
            }
        }

        // Final cross-lane reduction and write (one writer per row half).
#pragma unroll
        for (int r = 0; r < 8; ++r) {
            float ls = lsum[r], x0 = ox[r], y0 = oy[r];
#pragma unroll
            for (int sh = 1; sh <= 8; sh <<= 1) {
                ls += __shfl_xor(ls, sh, 32);
                x0 += __shfl_xor(x0, sh, 32);
                y0 += __shfl_xor(y0, sh, 32);
            }
            if (lrow == 0) {
                const int   qrow = qt * 16 + r + 8 * lhalf;
                const float inv  = 1.0f / ls;
                Op[(size_t)qrow * D_DIM + 0] = x0 * inv;
                Op[(size_t)qrow * D_DIM + 1] = y0 * inv;
            }
        }
    }
}

// ---------------------------------------------------------------------------
// Kernel 3: output projection.  Y = O @ Wo^T  -> d_out.
// ---------------------------------------------------------------------------
__global__ __launch_bounds__(256)
void proj_kernel(const float* __restrict__ X,
                 const float* __restrict__ W,
                 float* __restrict__ Y) {
    const int lane  = threadIdx.x & 31;
    const int wid   = (blockIdx.x * blockDim.x + threadIdx.x) >> 5;
    const int mtile = wid >> 3;
    const int ntile = wid & 7;
    const int lrow  = lane & 15;
    const int lhalf = lane >> 4;

    const float* xp = X + (size_t)(mtile * 16 + lrow) * D_DIM + 2 * lhalf;
    const float* wp = W + (size_t)(ntile * 16 + lrow) * D_DIM + 2 * lhalf;

    v8f c = {};
#pragma unroll 4
    for (int k = 0; k < D_DIM; k += 4) {
        v2f a = *(const v2f*)(xp + k);
        v2f b = *(const v2f*)(wp + k);
        c = wmma_f32_16x16x4(a, b, c);
    }

    const int row0 = mtile * 16 + 8 * lhalf;
    const int col  = ntile * 16 + lrow;
#pragma unroll
    for (int r = 0; r < 8; ++r)
        Y[(size_t)(row0 + r) * D_DIM + col] = c[r];
}

// ---------------------------------------------------------------------------
extern "C" void kernel_launch(void* const* d_in, const int* in_sizes, int n_in,
                              void* d_out, int out_size, void* d_ws, size_t ws_size,
                              hipStream_t stream) {
    const float* x  = (const float*)d_in[0];
    const float* Wq = (const float*)d_in[1];
    const float* Wk = (const float*)d_in[2];
    const float* Wv = (const float*)d_in[3];
    const float* Wo = (const float*)d_in[4];
    float* out = (float*)d_out;

    const size_t plane = (size_t)NROWS * D_DIM;  // 2048*128 floats = 1 MB
    float* Q = (float*)d_ws;
    float* K = Q + plane;
    float* V = K + plane;
    float* O = V + plane;

    // 1024 wave-tiles (128 M x 8 N) at 8 waves/block -> 128 blocks.
    qkv_kernel<<<128, 256, 0, stream>>>(x, Wq, Wk, Wv, Q, K, V);
    // one block per (b,h) pair: 2*64 = 128 blocks.
    attn_kernel<<<128, 256, 0, stream>>>(Q, K, V, O);
    proj_kernel<<<128, 256, 0, stream>>>(O, Wo, out);
}